// GraphSage_6193342841339
// MI455X (gfx1250) — compile-verified
//
#include <hip/hip_runtime.h>

typedef __attribute__((ext_vector_type(2))) float v2f;
typedef __attribute__((ext_vector_type(8))) float v8f;

#define FEAT      128   // D == OUT == 128
#define KDIM      256   // 2*FEAT (concat of self || agg)
#define KNB       10    // sampled neighbors
#define LDSS      (KDIM + 4)  // padded LDS row stride (260 dwords): lane m -> bank 4m, conflict-free

// One fused SAGE layer: out[m, :] = relu( concat(feats[nodes[m]], mean_j feats[neigh[m,j]]) @ W^T )
// Block = 256 threads = 8 waves; each block owns a 16-row M tile; each wave owns a 16-col N tile.
__global__ void __launch_bounds__(256)
sage_layer_fused(const float* __restrict__ feats,  // [*, 128] feature table
                 const int*   __restrict__ nodes,  // [M]
                 const int*   __restrict__ neigh,  // [M, 10]
                 const float* __restrict__ W,      // [128, 256] row-major (out, k)
                 float*       __restrict__ out,    // [M, 128]
                 int M) {
  __shared__ float sA[16 * LDSS];   // 16 x 256 A-tile, padded

  const int tid  = threadIdx.x;
  const int wave = tid >> 5;        // 0..7
  const int lane = tid & 31;
  const int m0   = blockIdx.x << 4; // 16 rows per block

  // ---- Stage 1: gather self + mean(neighbors) into LDS (2 rows per wave) ----
  #pragma unroll
  for (int rr = 0; rr < 2; ++rr) {
    const int r   = (wave << 1) + rr;   // wave-uniform -> index loads scalarize
    const int row = m0 + r;
    const float4 sv = ((const float4*)(feats + (size_t)nodes[row] * FEAT))[lane];
    float4 acc = make_float4(0.f, 0.f, 0.f, 0.f);
    #pragma unroll
    for (int j = 0; j < KNB; ++j) {
      const float4 nv = ((const float4*)(feats + (size_t)neigh[row * KNB + j] * FEAT))[lane];
      acc.x += nv.x; acc.y += nv.y; acc.z += nv.z; acc.w += nv.w;
    }
    const float inv = 1.0f / (float)KNB;
    *(float4*)&sA[r * LDSS + (lane << 2)]        = sv;                       // self -> k[0,128)
    *(float4*)&sA[r * LDSS + FEAT + (lane << 2)] =
        make_float4(acc.x * inv, acc.y * inv, acc.z * inv, acc.w * inv);     // agg  -> k[128,256)
  }
  __syncthreads();

  // ---- Stage 2: 16x16 tile GEMM via V_WMMA_F32_16X16X4_F32, K-loop over 256 ----
  // A frag (16x4 f32): lanes 0-15 -> M=lane, {K0,K1}; lanes 16-31 -> M=lane-16, {K2,K3}
  // B frag (4x16 f32): lanes 0-15 -> N=lane, {K0,K1}; lanes 16-31 -> N=lane-16, {K2,K3}
  const int n0    = wave << 4;
  const int mloc  = lane & 15;
  const int nloc  = lane & 15;
  const int khalf = (lane >> 4) << 1;  // 0 or 2

  const float* __restrict__ arow = &sA[mloc * LDSS + khalf];
  const float* __restrict__ wrow = W + (size_t)(n0 + nloc) * KDIM + khalf;   // B[k,n] = W[n,k]

  v8f c = {};
  #pragma unroll 8
  for (int k0 = 0; k0 < KDIM; k0 += 4) {
    v2f a = *(const v2f*)(arow + k0);
    v2f b = *(const v2f*)(wrow + k0);
    c = __builtin_amdgcn_wmma_f32_16x16x4_f32(
        /*neg_a=*/false, a, /*neg_b=*/false, b,
        /*c_mod=*/(short)0, c, /*reuse_a=*/false, /*reuse_b=*/false);
  }

  // ---- Stage 3: ReLU + store. C/D layout: VGPR v -> M = v (lanes 0-15) / v+8 (lanes 16-31)
  const int mh = (lane >> 4) << 3;
  #pragma unroll
  for (int v = 0; v < 8; ++v) {
    float x = c[v];
    x = x > 0.0f ? x : 0.0f;
    out[(size_t)(m0 + mh + v) * FEAT + n0 + nloc] = x;
  }
}

extern "C" void kernel_launch(void* const* d_in, const int* in_sizes, int n_in,
                              void* d_out, int out_size, void* d_ws, size_t ws_size,
                              hipStream_t stream) {
  const float* raw    = (const float*)d_in[0];  // [N, 128]
  const float* W1     = (const float*)d_in[1];  // [128, 256]
  const float* W2     = (const float*)d_in[2];  // [128, 256]
  const int*   nodes1 = (const int*)d_in[3];    // [B1]
  const int*   neigh1 = (const int*)d_in[4];    // [B1, 10]
  const int*   nodes2 = (const int*)d_in[5];    // [B2]
  const int*   neigh2 = (const int*)d_in[6];    // [B2, 10]

  const int B1 = in_sizes[3];  // 40960
  const int B2 = in_sizes[5];  // 4096

  float* h1 = (float*)d_ws;    // [B1, 128] hidden table (21 MB)

  // Layer 1: raw features -> h1
  sage_layer_fused<<<B1 / 16, 256, 0, stream>>>(raw, nodes1, neigh1, W1, h1, B1);
  // Layer 2: h1 (local indices) -> output
  sage_layer_fused<<<B2 / 16, 256, 0, stream>>>(h1, nodes2, neigh2, W2, (float*)d_out, B2);
}